// QuLinear_60129542144056
// MI455X (gfx1250) — compile-verified
//
#include <hip/hip_runtime.h>
#include <math.h>

// ---------------------------------------------------------------------------
// Problem constants (match reference)
// ---------------------------------------------------------------------------
static constexpr int NQ    = 10;
static constexpr int DIM   = 1 << NQ;      // 1024
static constexpr int BATCH = 8192;
static constexpr int NL    = 4;            // layers

typedef float v2f __attribute__((ext_vector_type(2)));
typedef float v8f __attribute__((ext_vector_type(8)));

// ---------------------------------------------------------------------------
// CDNA5 async global->LDS copy (GLOBAL_LOAD_ASYNC_TO_LDS_B128, ASYNCcnt)
// via inline asm.
//   vdst = 32-bit LDS byte offset (low 32 bits of generic shared pointer)
//   vaddr = 64-bit global address (VGPR pair)
// ---------------------------------------------------------------------------
#if defined(__gfx1250__)
#define USE_ASYNC_LDS 1
#else
#define USE_ASYNC_LDS 0
#endif

__device__ __forceinline__ void copy16_g2l(const float* g, float* l) {
#if USE_ASYNC_LDS
  const unsigned lds_off = (unsigned)(size_t)(void*)l;   // addr[31:0] = LDS offset
  const unsigned long long gaddr = (unsigned long long)(size_t)(const void*)g;
  asm volatile("global_load_async_to_lds_b128 %0, %1, off"
               :
               : "v"(lds_off), "v"(gaddr)
               : "memory");
#else
  *reinterpret_cast<float4*>(l) = *reinterpret_cast<const float4*>(g);
#endif
}

__device__ __forceinline__ void wait_copies_le4() {
#if USE_ASYNC_LDS
  asm volatile("s_wait_asynccnt 4" ::: "memory");
#endif
}
__device__ __forceinline__ void wait_copies_le0() {
#if USE_ASYNC_LDS
  asm volatile("s_wait_asynccnt 0" ::: "memory");
#endif
}

// ---------------------------------------------------------------------------
// Kernel 1: per-row inverse norm (one wave32 per row)
// ---------------------------------------------------------------------------
__global__ __launch_bounds__(256) void row_invnorm_kernel(
    const float* __restrict__ x, float* __restrict__ invnorm) {
  const int wave = threadIdx.x >> 5;
  const int lane = threadIdx.x & 31;
  const int row  = blockIdx.x * 8 + wave;
  const float* xr = x + (size_t)row * DIM;
  float s = 0.f;
#pragma unroll 8
  for (int i = lane; i < DIM; i += 32) {
    float v = xr[i];
    s += v * v;
  }
#pragma unroll
  for (int off = 16; off > 0; off >>= 1) s += __shfl_xor(s, off, 32);
  if (lane == 0) invnorm[row] = 1.0f / fmaxf(sqrtf(s), 1e-12f);
}

// ---------------------------------------------------------------------------
// Kernel 2: simulate circuit on basis state e_k (one column per block) and
// store Re(U e_k) K-pair-interleaved:
//     Mtp[(k>>1)*2*DIM + 2*j + (k&1)] = Re(U[j,k])
// so a 16B chunk of Mtp is exactly a WMMA B-fragment pair for 2 cols, and the
// async B128 fill lands the pair-interleaved LDS layout with no rearrange.
// ---------------------------------------------------------------------------
__device__ __forceinline__ void pair_idx(int t, int p, int& i0, int& i1) {
  const int mask = (1 << p) - 1;
  i0 = ((t & ~mask) << 1) | (t & mask);
  i1 = i0 | (1 << p);
}

__global__ __launch_bounds__(512) void build_unitary_kernel(
    const float* __restrict__ w, float* __restrict__ Mtp) {
  __shared__ float re[DIM];
  __shared__ float im[DIM];
  const int t = threadIdx.x;   // 0..511
  const int k = blockIdx.x;    // basis column

  re[t]       = (t == k) ? 1.f : 0.f;
  re[t + 512] = ((t + 512) == k) ? 1.f : 0.f;
  im[t]       = 0.f;
  im[t + 512] = 0.f;
  __syncthreads();

  // entangling layers: RX sweep, RY sweep, RZ sweep, CNOT ring
  for (int l = 0; l < NL; ++l) {
    const float* wl = w + l * 3 * NQ;
    for (int g = 0; g < 3; ++g) {          // 0=RX, 1=RY, 2=RZ
      for (int q = 0; q < NQ; ++q) {
        const float th = wl[g * NQ + q];
        const float c = cosf(0.5f * th);
        const float s = sinf(0.5f * th);
        int i0, i1;
        pair_idx(t, 9 - q, i0, i1);
        const float r0 = re[i0], m0 = im[i0];
        const float r1 = re[i1], m1 = im[i1];
        float nr0, nm0, nr1, nm1;
        if (g == 0) {        // RX: [[c,-is],[-is,c]]
          nr0 = c * r0 + s * m1;  nm0 = c * m0 - s * r1;
          nr1 = c * r1 + s * m0;  nm1 = c * m1 - s * r0;
        } else if (g == 1) { // RY: [[c,-s],[s,c]]
          nr0 = c * r0 - s * r1;  nm0 = c * m0 - s * m1;
          nr1 = s * r0 + c * r1;  nm1 = s * m0 + c * m1;
        } else {             // RZ: diag(e^{-i th/2}, e^{+i th/2})
          nr0 = c * r0 + s * m0;  nm0 = c * m0 - s * r0;
          nr1 = c * r1 - s * m1;  nm1 = c * m1 + s * r1;
        }
        re[i0] = nr0; im[i0] = nm0;
        re[i1] = nr1; im[i1] = nm1;
        __syncthreads();
      }
    }
    // CNOT ring: ctrl q -> tgt (q+1)%NQ ; 256 swaps handled by t<256
    for (int q = 0; q < NQ; ++q) {
      const int pc = 9 - q;
      const int pt = 9 - ((q + 1) % NQ);
      if (t < 256) {
        const int plo = (pc < pt) ? pc : pt;
        const int phi = (pc < pt) ? pt : pc;
        const int mlo = (1 << plo) - 1;
        const int mhi = (1 << phi) - 1;
        int i = ((t & ~mlo) << 1) | (t & mlo);  // insert 0 at plo
        i = ((i & ~mhi) << 1) | (i & mhi);      // insert 0 at phi
        const int iA = i | (1 << pc);           // ctrl=1, tgt=0
        const int iB = iA | (1 << pt);          // ctrl=1, tgt=1
        const float tr = re[iA], tm = im[iA];
        re[iA] = re[iB]; im[iA] = im[iB];
        re[iB] = tr;     im[iB] = tm;
      }
      __syncthreads();
    }
  }

  // final rotations: RY, RZ, RY
  {
    const float* wf = w + NL * 3 * NQ;
    for (int g = 0; g < 3; ++g) {
      const int type = (g == 1) ? 2 : 1;   // 1=RY, 2=RZ
      for (int q = 0; q < NQ; ++q) {
        const float th = wf[g * NQ + q];
        const float c = cosf(0.5f * th);
        const float s = sinf(0.5f * th);
        int i0, i1;
        pair_idx(t, 9 - q, i0, i1);
        const float r0 = re[i0], m0 = im[i0];
        const float r1 = re[i1], m1 = im[i1];
        float nr0, nm0, nr1, nm1;
        if (type == 1) {     // RY
          nr0 = c * r0 - s * r1;  nm0 = c * m0 - s * m1;
          nr1 = s * r0 + c * r1;  nm1 = s * m0 + c * m1;
        } else {             // RZ
          nr0 = c * r0 + s * m0;  nm0 = c * m0 - s * r0;
          nr1 = c * r1 - s * m1;  nm1 = c * m1 + s * r1;
        }
        re[i0] = nr0; im[i0] = nm0;
        re[i1] = nr1; im[i1] = nm1;
        __syncthreads();
      }
    }
  }

  // pair-interleaved store
  const size_t base = (size_t)(k >> 1) * (2 * DIM) + (k & 1);
  Mtp[base + 2 * t]         = re[t];
  Mtp[base + 2 * (t + 512)] = re[t + 512];
}

// ---------------------------------------------------------------------------
// Kernel 3: Y = (X @ Mt) * diag(invnorm) via V_WMMA_F32_16X16X4_F32.
// Block = 256 threads (8 waves). C tile 128x128; wave w owns rows
// [w*16,w*16+16) x 128 cols -> 8 accumulators of 16x16 f32.
// K-chunks of 16, double-buffered in LDS, async global->LDS fills
// (4 x B128 per thread per chunk) overlapped with WMMA compute.
// B is K-pair-interleaved in both global and LDS: fragment = one v2f load.
// ---------------------------------------------------------------------------
static constexpr int KC   = 16;
static constexpr int KP   = KC / 2;  // pair-rows per chunk
static constexpr int LDA  = 20;      // 16 + pad (f32)  -> conflict-free A reads
static constexpr int LDBP = 288;     // 256 + 32 pad -> half-waves on disjoint banks

__global__ __launch_bounds__(256) void gemm_wmma_kernel(
    const float* __restrict__ X, const float* __restrict__ Mtp,
    const float* __restrict__ invnorm, float* __restrict__ Y) {
  __shared__ float As[2][128 * LDA];
  __shared__ float Bs[2][KP * LDBP];

  const int t    = threadIdx.x;
  const int lane = t & 31;
  const int wave = t >> 5;
  const int bm   = blockIdx.y * 128;
  const int bn   = blockIdx.x * 128;

  v8f acc[8] = {};

  // copy assignments:
  // A tile 128xKC        : thread -> row t>>1, 8 floats at col (t&1)*8  (2 x B128)
  // B tile KP x 256 (pi) : thread -> pair-row t>>5, 8 floats at (t&31)*8 (2 x B128)
  const int rowA = t >> 1, colA = (t & 1) * 8;
  const int prB  = t >> 5, foB  = (lane) * 8;

  const int m  = lane & 15;          // fragment row/col within 16x16
  const int kh = (lane >> 4) * 2;    // K sub-pair per ISA 16x4 A layout

  const float* gA0 = X + (size_t)(bm + rowA) * DIM + colA;            // + kc
  const float* gB0 = Mtp + (size_t)prB * (2 * DIM) + bn * 2 + foB;    // + (kc/2)*2*DIM

  // ---- prologue: start chunk 0 into stage 0
  {
    copy16_g2l(gA0 + 0, &As[0][rowA * LDA + colA]);
    copy16_g2l(gA0 + 4, &As[0][rowA * LDA + colA + 4]);
    copy16_g2l(gB0 + 0, &Bs[0][prB * LDBP + foB]);
    copy16_g2l(gB0 + 4, &Bs[0][prB * LDBP + foB + 4]);
  }

  int st = 0;
  for (int kc = 0; kc < DIM; kc += KC, st ^= 1) {
    const bool has_next = (kc + KC) < DIM;
    if (has_next) {
      const int nx = st ^ 1;
      const float* gA = gA0 + (kc + KC);
      const float* gB = gB0 + (size_t)((kc + KC) >> 1) * (2 * DIM);
      copy16_g2l(gA + 0, &As[nx][rowA * LDA + colA]);
      copy16_g2l(gA + 4, &As[nx][rowA * LDA + colA + 4]);
      copy16_g2l(gB + 0, &Bs[nx][prB * LDBP + foB]);
      copy16_g2l(gB + 4, &Bs[nx][prB * LDBP + foB + 4]);
      wait_copies_le4();   // retire current chunk's 4 copies (in-order)
    } else {
      wait_copies_le0();
    }
    __syncthreads();

    const float* __restrict__ Asl = As[st];
    const float* __restrict__ Bsl = Bs[st];
#pragma unroll
    for (int kk = 0; kk < KC; kk += 4) {
      // A fragment: {A[row][k0], A[row][k0+1]}, k0 = kk+kh even -> aligned v2f
      const v2f a =
          *reinterpret_cast<const v2f*>(&Asl[(wave * 16 + m) * LDA + kk + kh]);
      const int prow = (kk + kh) >> 1;
#pragma unroll
      for (int n = 0; n < 8; ++n) {
        // B fragment: {B[k0][col], B[k0+1][col]} -> one aligned v2f load
        const v2f b = *reinterpret_cast<const v2f*>(
            &Bsl[prow * LDBP + (n * 16 + m) * 2]);
        // (neg_a, A, neg_b, B, c_mod, C, reuse_a, reuse_b)
        acc[n] = __builtin_amdgcn_wmma_f32_16x16x4_f32(
            false, a, false, b, (short)0, acc[n], false, false);
      }
    }
    __syncthreads();   // all waves done reading stage st before it is refilled
  }

  // Epilogue: C/D layout — VGPR v: lanes 0-15 -> M=v, lanes 16-31 -> M=v+8.
  // Fold the 1/||x|| row scale here (linear, identical to scaling X).
  const int rbase = bm + wave * 16 + ((lane >> 4) << 3);
#pragma unroll
  for (int v = 0; v < 8; ++v) {
    const int row = rbase + v;
    const float sc = invnorm[row];
#pragma unroll
    for (int n = 0; n < 8; ++n) {
      Y[(size_t)row * DIM + bn + n * 16 + m] = acc[n][v] * sc;
    }
  }
}

// ---------------------------------------------------------------------------
// Launch
// ---------------------------------------------------------------------------
extern "C" void kernel_launch(void* const* d_in, const int* in_sizes, int n_in,
                              void* d_out, int out_size, void* d_ws, size_t ws_size,
                              hipStream_t stream) {
  const float* x      = (const float*)d_in[0];   // [8192,1024]
  const float* weight = (const float*)d_in[1];   // [150]
  float* out = (float*)d_out;                    // [8192,1,1024]

  float* invnorm = (float*)d_ws;                 // 8192 f32
  float* Mtp     = invnorm + BATCH;              // 1024*1024 f32, pair-interleaved

  row_invnorm_kernel<<<BATCH / 8, 256, 0, stream>>>(x, invnorm);
  build_unitary_kernel<<<DIM, 512, 0, stream>>>(weight, Mtp);
  gemm_wmma_kernel<<<dim3(DIM / 128, BATCH / 128), 256, 0, stream>>>(
      x, Mtp, invnorm, out);
}